// HeatLayer_69638599737397
// MI455X (gfx1250) — compile-verified
//
#include <hip/hip_runtime.h>
#include <hip/hip_bf16.h>

typedef __attribute__((ext_vector_type(2))) float v2f;
typedef __attribute__((ext_vector_type(8))) float v8f;

#define S_TOT   8192
#define D_TOT   768
#define CHUNK   64                 // pass-3 chunk (rows per scan block)
#define SUB     32                 // pass-1 sub-chunk (rows per partial block)
#define NSUB    (S_TOT / SUB)      // 256
#define NCHUNK  (S_TOT / CHUNK)    // 128

// ---------------------------------------------------------------------------
// Pass 1: per-sub-chunk partial sums (sumH, sum relu(h)*exp(-beta*t)).
// One thread per 4-column quad, float4 loads -> 4x fewer VMEM instructions.
// grid = NSUB blocks of 192 threads (192*4 = 768 columns).
// ---------------------------------------------------------------------------
__global__ void heat_pass1(const float* __restrict__ h,
                           const float* __restrict__ t,
                           const float* __restrict__ beta_p,
                           float* __restrict__ partial) {
    __shared__ float Wl[SUB];                   // exp(-beta * t[row])
    const int sub  = blockIdx.x;
    const int q    = threadIdx.x;               // column quad 0..191
    const int row0 = sub * SUB;
    const float beta = *beta_p;

    if (q < SUB) Wl[q] = __expf(-beta * t[row0 + q]);
    __syncthreads();

    const float4* h4 = (const float4*)(h + (size_t)row0 * D_TOT) + q;
    float4 sH = make_float4(0.f, 0.f, 0.f, 0.f);
    float4 sR = make_float4(0.f, 0.f, 0.f, 0.f);
#pragma unroll 4
    for (int r = 0; r < SUB; ++r) {
        const float4 x = h4[(size_t)r * (D_TOT / 4)];
        const float  w = Wl[r];
        sH.x += x.x; sH.y += x.y; sH.z += x.z; sH.w += x.w;
        sR.x += fmaxf(x.x, 0.f) * w;
        sR.y += fmaxf(x.y, 0.f) * w;
        sR.z += fmaxf(x.z, 0.f) * w;
        sR.w += fmaxf(x.w, 0.f) * w;
    }
    // partial layout [sub][col][2]: 8 contiguous floats per quad -> two b128 stores
    float4* p4 = (float4*)(partial + ((size_t)sub * D_TOT + (size_t)q * 4) * 2);
    p4[0] = make_float4(sH.x, sR.x, sH.y, sR.y);
    p4[1] = make_float4(sH.z, sR.z, sH.w, sR.w);
}

// ---------------------------------------------------------------------------
// Pass 2: in-place exclusive scan over the NSUB sub-chunk partials.
// 1536 independent serial scans of length 256 (one thread each).
// ---------------------------------------------------------------------------
__global__ void heat_pass2(float* __restrict__ partial) {
    const int tid = blockIdx.x * blockDim.x + threadIdx.x;  // (col, quantity)
    if (tid >= D_TOT * 2) return;
    const size_t stride = (size_t)D_TOT * 2;
    float acc = 0.0f;
    for (int c = 0; c < NSUB; ++c) {
        const size_t k = (size_t)c * stride + tid;
        const float v = partial[k];
        partial[k] = acc;                                   // exclusive prefix
        acc += v;
    }
}

// ---------------------------------------------------------------------------
// Pass 3: WMMA scan. Each wave owns a 16-column strip of one 64-row chunk and
// scans it in 16-row tiles: cum = L @ X via 4 chained V_WMMA_F32_16X16X4_F32
// per quantity (L = 16x16 lower-triangular ones, sliced into 16x4 K-panels).
// Inter-tile carry broadcast with ds_bpermute_b32. Combine + store.
// grid = (NCHUNK, D/128), block = 256 (8 waves).
// ---------------------------------------------------------------------------
__global__ void heat_pass3(const float* __restrict__ h,
                           const float* __restrict__ t,
                           const float* __restrict__ off,   // exclusive sub offsets
                           const float* __restrict__ eps_p,
                           const float* __restrict__ beta_p,
                           float* __restrict__ out) {
    __shared__ float Wl[CHUNK];   // exp(-beta * t[row])
    __shared__ float Fl[CHUNK];   // eps * exp(+beta * t[row])

    const int chunk = blockIdx.x;
    const int tidx  = threadIdx.x;
    const int wave  = tidx >> 5;
    const int lane  = tidx & 31;
    const int lo    = lane & 15;          // N / M-low index
    const int hi    = lane >> 4;          // lane-half selector
    const int row0  = chunk * CHUNK;
    const float eps  = *eps_p;
    const float beta = *beta_p;

    if (tidx < CHUNK) {
        const float tv = t[row0 + tidx];
        Wl[tidx] = __expf(-beta * tv);
        Fl[tidx] = eps * __expf(beta * tv);
    }
    __syncthreads();

    const int col = blockIdx.y * 128 + wave * 16 + lo;

    // carries = exclusive prefix over rows [0, row0): sub-chunk index 2*chunk
    float carryH = off[((size_t)(2 * chunk) * D_TOT + col) * 2 + 0];
    float carryR = off[((size_t)(2 * chunk) * D_TOT + col) * 2 + 1];

    // A-panels: L[m, 4s + kl], m = lane&15, kl = reg + 2*(lane>=16)  (ISA 16x4 layout)
    v2f A[4];
#pragma unroll
    for (int s = 0; s < 4; ++s) {
        A[s][0] = (4 * s + 0 + 2 * hi <= lo) ? 1.0f : 0.0f;
        A[s][1] = (4 * s + 1 + 2 * hi <= lo) ? 1.0f : 0.0f;
    }

#pragma unroll
    for (int tile = 0; tile < CHUNK / 16; ++tile) {
        const int tbase = tile * 16;

        // B-panels: B_s[kl, n] = X[tbase + 4s + kl, col], kl = reg + 2*hi
        v2f BH[4], BR[4];
#pragma unroll
        for (int s = 0; s < 4; ++s) {
#pragma unroll
            for (int j = 0; j < 2; ++j) {
                const int lr = tbase + 4 * s + j + 2 * hi;   // local row
                const float x = h[(size_t)(row0 + lr) * D_TOT + col];
                BH[s][j] = x;
                BR[s][j] = fmaxf(x, 0.0f) * Wl[lr];
            }
        }

        // C init = carry replicated; every C lane maps to column (lane&15)
        v8f CH, CR;
#pragma unroll
        for (int r = 0; r < 8; ++r) { CH[r] = carryH; CR[r] = carryR; }

#pragma unroll
        for (int s = 0; s < 4; ++s) {
            CH = __builtin_amdgcn_wmma_f32_16x16x4_f32(
                     false, A[s], false, BH[s], (short)0, CH, false, false);
            CR = __builtin_amdgcn_wmma_f32_16x16x4_f32(
                     false, A[s], false, BR[s], (short)0, CR, false, false);
        }

        // carry update: row M=15 is D-VGPR 7, lanes 16..31 (N = lane-16)
        const int pidx = (16 + lo) * 4;
        carryH = __uint_as_float(
            (unsigned)__builtin_amdgcn_ds_bpermute(pidx, (int)__float_as_uint(CH[7])));
        carryR = __uint_as_float(
            (unsigned)__builtin_amdgcn_ds_bpermute(pidx, (int)__float_as_uint(CR[7])));

        // combine + store: C reg r holds row tbase + r + 8*hi, column col
#pragma unroll
        for (int r = 0; r < 8; ++r) {
            const int lr = tbase + r + 8 * hi;
            out[(size_t)(row0 + lr) * D_TOT + col] = CH[r] + Fl[lr] * CR[r];
        }
    }
}

// ---------------------------------------------------------------------------
extern "C" void kernel_launch(void* const* d_in, const int* in_sizes, int n_in,
                              void* d_out, int out_size, void* d_ws, size_t ws_size,
                              hipStream_t stream) {
    const float* h      = (const float*)d_in[0];   // [1, 8192, 768] f32
    const float* t      = (const float*)d_in[1];   // [1, 8192]      f32
    const float* eps_p  = (const float*)d_in[2];   // scalar
    const float* beta_p = (const float*)d_in[3];   // scalar
    float* out     = (float*)d_out;                // [1, 8192, 768] f32
    float* partial = (float*)d_ws;                 // [NSUB][D][2] f32 = 1.57 MB

    heat_pass1<<<NSUB, D_TOT / 4, 0, stream>>>(h, t, beta_p, partial);

    heat_pass2<<<(D_TOT * 2 + 255) / 256, 256, 0, stream>>>(partial);

    dim3 g3(NCHUNK, D_TOT / 128);
    heat_pass3<<<g3, 256, 0, stream>>>(h, t, partial, eps_p, beta_p, out);
}